// TimeUserPair_62577673503501
// MI455X (gfx1250) — compile-verified
//
#include <hip/hip_runtime.h>
#include <math.h>

// Problem constants (match reference)
#define Bsz   32
#define Ssz   2048
#define Dsz   512
#define Hsz   4
#define HDsz  128
#define Tsz   24
#define Nsz   (Bsz*Ssz)       // 65536
#define Gsz   (Hsz*Tsz)       // 96 = flattened (head, timeslot)

// Workspace layout (float offsets)
#define WS_KMAT 0                                  // [H][T][HD]    = 12288
#define WS_VMAT (WS_KMAT + Hsz*Tsz*HDsz)           // [H][T][HD]    = 12288
#define WS_QU   (WS_VMAT + Hsz*Tsz*HDsz)           // [B][H][HD]    = 16384
#define WS_QT   (WS_QU   + Bsz*Hsz*HDsz)           // [T][H][HD]    = 12288
#define WS_SU   (WS_QT   + Tsz*Hsz*HDsz)           // [B][H][T]     = 3072
#define WS_ST   (WS_SU   + Bsz*Hsz*Tsz)            // [T][H][T]     = 2304
#define WS_VWU  (WS_ST   + Tsz*Hsz*Tsz)            // [96][512]     = 49152
#define WS_VWT  (WS_VWU  + Gsz*Dsz)                // [96][24]      = 2304

typedef __attribute__((ext_vector_type(2))) float v2f;
typedef __attribute__((ext_vector_type(4))) float f4;
typedef __attribute__((ext_vector_type(8))) float v8f;
typedef __attribute__((ext_vector_type(4))) int   i4;

// ---------------- Precompute 1: k[h,t,:], v[h,t,:] -----------------
__global__ __launch_bounds__(128)
void kv_kernel(const float* __restrict__ sm, const float* __restrict__ Wk,
               const float* __restrict__ bk, const float* __restrict__ Wv,
               const float* __restrict__ bv, float* __restrict__ ws) {
  int g = blockIdx.x;            // h*T + t
  int h = g / Tsz, t = g % Tsz;
  int e = threadIdx.x;           // 0..127
  const float* srow = sm + (size_t)t * Dsz;
  const float* wk = Wk + (size_t)h * Dsz * HDsz + e;
  const float* wv = Wv + (size_t)h * Dsz * HDsz + e;
  float ak = 0.f, av = 0.f;
  for (int d = 0; d < Dsz; ++d) {
    float s = srow[d];
    ak = fmaf(s, wk[(size_t)d * HDsz], ak);
    av = fmaf(s, wv[(size_t)d * HDsz], av);
  }
  ws[WS_KMAT + (size_t)g * HDsz + e] = ak + bk[h * HDsz + e];
  ws[WS_VMAT + (size_t)g * HDsz + e] = av + bv[h * HDsz + e];
}

// ---------------- Precompute 2: qU[b,h,:] (incl. bq), qT[tau,h,:] ----
__global__ __launch_bounds__(128)
void q_kernel(const float* __restrict__ timeslot, const float* __restrict__ upref,
              const int* __restrict__ user_x, const float* __restrict__ Wq,
              const float* __restrict__ bq, float* __restrict__ ws) {
  int g = blockIdx.x;
  int e = threadIdx.x;
  if (g < Bsz * Hsz) {
    int b = g / Hsz, h = g % Hsz;
    const float* row = upref + (size_t)user_x[b] * Dsz;
    const float* w = Wq + (size_t)h * (2 * Dsz) * HDsz + e;   // rows 0..D-1
    float acc = 0.f;
    for (int d = 0; d < Dsz; ++d) acc = fmaf(row[d], w[(size_t)d * HDsz], acc);
    ws[WS_QU + (size_t)g * HDsz + e] = acc + bq[h * HDsz + e];
  } else {
    int j = g - Bsz * Hsz;
    int tau = j / Hsz, h = j % Hsz;
    const float* row = timeslot + (size_t)tau * Dsz;
    const float* w = Wq + (size_t)h * (2 * Dsz) * HDsz + (size_t)Dsz * HDsz + e; // rows D..2D-1
    float acc = 0.f;
    for (int d = 0; d < Dsz; ++d) acc = fmaf(row[d], w[(size_t)d * HDsz], acc);
    ws[WS_QT + (size_t)j * HDsz + e] = acc;
  }
}

// ---------------- Precompute 3: SU[b,h,t], ST[tau,h,t] ---------------
__global__ __launch_bounds__(256)
void score_kernel(float* __restrict__ ws) {
  const float scale = 0.088388347648318447f;   // 1/sqrt(128)
  int i = blockIdx.x * 256 + threadIdx.x;
  int nSU = Bsz * Hsz * Tsz;                   // 3072
  int nST = Tsz * Hsz * Tsz;                   // 2304
  if (i < nSU) {
    int b = i / (Hsz * Tsz);
    int r = i % (Hsz * Tsz);
    int h = r / Tsz, t = r % Tsz;
    const float* q = ws + WS_QU + (size_t)(b * Hsz + h) * HDsz;
    const float* k = ws + WS_KMAT + (size_t)(h * Tsz + t) * HDsz;
    float acc = 0.f;
    for (int e = 0; e < HDsz; ++e) acc = fmaf(q[e], k[e], acc);
    ws[WS_SU + i] = acc * scale;
  } else if (i < nSU + nST) {
    int j = i - nSU;
    int tau = j / (Hsz * Tsz);
    int r = j % (Hsz * Tsz);
    int h = r / Tsz, t = r % Tsz;
    const float* q = ws + WS_QT + (size_t)(tau * Hsz + h) * HDsz;
    const float* k = ws + WS_KMAT + (size_t)(h * Tsz + t) * HDsz;
    float acc = 0.f;
    for (int e = 0; e < HDsz; ++e) acc = fmaf(q[e], k[e], acc);
    ws[WS_ST + j] = acc * scale;
  }
}

// ---------------- Precompute 4: VWu[96,512], VWt[96,24] --------------
__global__ __launch_bounds__(256)
void vw_kernel(const float* __restrict__ Wu, const float* __restrict__ Wt,
               float* __restrict__ ws) {
  __shared__ float vsh[HDsz];
  int g = blockIdx.x;            // h*T + t
  int h = g / Tsz;
  int tid = threadIdx.x;
  if (tid < HDsz) vsh[tid] = ws[WS_VMAT + (size_t)g * HDsz + tid];
  __syncthreads();
  for (int c = tid; c < Dsz; c += 256) {
    float acc = 0.f;
    for (int e = 0; e < HDsz; ++e)
      acc = fmaf(vsh[e], Wu[(size_t)(h * HDsz + e) * Dsz + c], acc);
    ws[WS_VWU + (size_t)g * Dsz + c] = acc;
  }
  if (tid < Tsz) {
    float acc = 0.f;
    for (int e = 0; e < HDsz; ++e)
      acc = fmaf(vsh[e], Wt[(size_t)(h * HDsz + e) * Tsz + tid], acc);
    ws[WS_VWT + (size_t)g * Tsz + tid] = acc;
  }
}

// ---------------- Fused main kernel ---------------------------------
// 256 threads = 8 waves; 64 rows of n per block; grid = N/64.
#define ROWS 64
#define APAD 100   // padded LDS stride (kills bank conflicts; 16B-aligned rows)

__global__ __launch_bounds__(256)
void fused_kernel(const int* __restrict__ hour_x, const int* __restrict__ hour_mask,
                  const float* __restrict__ osc_u, const float* __restrict__ osc_v,
                  const float* __restrict__ bu, const float* __restrict__ bt,
                  const float* __restrict__ ws,
                  float* __restrict__ at_out, float* __restrict__ tl_out) {
  __shared__ float attn[ROWS][APAD];
  const int n0 = blockIdx.x * ROWS;
  const int tid = threadIdx.x;

  // ---- Phase 1: scores -> z -> softmax; one thread per (row, head).
  // Mapping h=tid>>6 keeps a wave's 32 lanes on consecutive n for one head:
  // osc reads become a contiguous 3 KB stream per wave, in b128 chunks.
  {
    int h = tid >> 6;            // 0..3
    int r = tid & 63;            // 0..63
    int n = n0 + r;
    int b = n / Ssz;
    int hour = hour_x[n];
    const f4* su4 = (const f4*)(ws + WS_SU + (size_t)(b * Hsz + h) * Tsz);
    const f4* st4 = (const f4*)(ws + WS_ST + (size_t)(hour * Hsz + h) * Tsz);
    const f4* ou4 = (const f4*)(osc_u + ((size_t)h * Nsz + n) * Tsz);
    const f4* ov4 = (const f4*)(osc_v + ((size_t)h * Nsz + n) * Tsz);
    const i4* mk4 = (const i4*)(hour_mask + (size_t)n * Tsz);
    f4 z[6];
    float m = -INFINITY;
#pragma unroll
    for (int j = 0; j < 6; ++j) {
      f4 s  = su4[j] + st4[j];
      i4 mm = mk4[j];
      f4 u  = __builtin_nontemporal_load(&ou4[j]);   // streamed once: TH=NT
      f4 v  = __builtin_nontemporal_load(&ov4[j]);
#pragma unroll
      for (int c = 0; c < 4; ++c) {
        float sc = (mm[c] == 1) ? -INFINITY : s[c];
        // (-inf)^2 -> +inf ; exp(-inf) -> 0 ; relu(-inf) -> 0  => z = 0 (matches jnp)
        float e  = __expf(-500.f * sc * sc);
        float zz = (u[c] - v[c]) * 0.01f * e + fmaxf(sc, 0.f);
        z[j][c] = zz;
        m = fmaxf(m, zz);
      }
    }
    float sum = 0.f;
#pragma unroll
    for (int j = 0; j < 6; ++j)
#pragma unroll
      for (int c = 0; c < 4; ++c) { float e = __expf(z[j][c] - m); z[j][c] = e; sum += e; }
    float inv = 1.f / sum;
    f4* arow = (f4*)&attn[r][h * Tsz];               // 16B-aligned (96h bytes, 400B rows)
#pragma unroll
    for (int j = 0; j < 6; ++j) arow[j] = z[j] * inv;
  }
  __syncthreads();

  // ---- Phase 2: at_emb[64,512] = attn[64,96] @ VWu[96,512] + bu, via f32 WMMA.
  // A fragments hoisted per row-tile (24 x v2f = 48 VGPRs), reused over 4 col-tiles.
  {
    const float* vwu = ws + WS_VWU;
    int w    = tid >> 5;         // wave id 0..7
    int lane = tid & 31;
    int hi   = lane >> 4;        // 0: K={0,1}; 1: K={2,3}
    int lr   = lane & 15;        // row (A) / col (B,D) within tile
    for (int rt = 0; rt < 4; ++rt) {
      int rowBase = rt * 16;
      v2f a[24];
#pragma unroll
      for (int kk = 0; kk < 24; ++kk) {
        int k0 = kk * 4 + hi * 2;
        a[kk].x = attn[rowBase + lr][k0];
        a[kk].y = attn[rowBase + lr][k0 + 1];
      }
      for (int ct = w; ct < 32; ct += 8) {
        int colBase = ct * 16;
        v8f acc = {};
#pragma unroll
        for (int kk = 0; kk < 24; ++kk) {           // K = 96 in steps of 4
          int k0 = kk * 4 + hi * 2;
          v2f bm;
          bm.x = vwu[(size_t)k0 * Dsz + colBase + lr];
          bm.y = vwu[(size_t)(k0 + 1) * Dsz + colBase + lr];
#if __has_builtin(__builtin_amdgcn_wmma_f32_16x16x4_f32)
          acc = __builtin_amdgcn_wmma_f32_16x16x4_f32(
                    false, a[kk], false, bm, (short)0, acc, false, false);
#else
#pragma unroll
          for (int rr = 0; rr < 8; ++rr)
            acc[rr] = fmaf(a[kk].x, bm.x, fmaf(a[kk].y, bm.y, acc[rr]));
#endif
        }
        int col = colBase + lr;
        float bias = bu[col];
#pragma unroll
        for (int rr = 0; rr < 8; ++rr) {
          int row = rowBase + rr + hi * 8;          // D layout: vgpr rr -> M=rr / M=8+rr
          __builtin_nontemporal_store(acc[rr] + bias,
                                      &at_out[(size_t)(n0 + row) * Dsz + col]);
        }
      }
    }
  }

  // ---- Phase 3: time_logits[64,24] = attn @ VWt[96,24] + bt ----
  {
    const float* vwt = ws + WS_VWT;
    for (int i = tid; i < ROWS * Tsz; i += 256) {
      int r = i / Tsz, t = i % Tsz;
      float acc = bt[t];
#pragma unroll 8
      for (int g = 0; g < Gsz; ++g)
        acc = fmaf(attn[r][g], vwt[g * Tsz + t], acc);
      __builtin_nontemporal_store(acc, &tl_out[(size_t)(n0 + r) * Tsz + t]);
    }
  }
}

extern "C" void kernel_launch(void* const* d_in, const int* in_sizes, int n_in,
                              void* d_out, int out_size, void* d_ws, size_t ws_size,
                              hipStream_t stream) {
  const float* timeslot = (const float*)d_in[0];
  const float* smoothed = (const float*)d_in[1];
  /* d_in[2] user_embedded: unused by reference */
  const int*   user_x   = (const int*)d_in[3];
  const int*   hour_x   = (const int*)d_in[4];
  const int*   hour_mask= (const int*)d_in[5];
  const float* upref    = (const float*)d_in[6];
  const float* Wq       = (const float*)d_in[7];
  const float* bq       = (const float*)d_in[8];
  const float* Wk       = (const float*)d_in[9];
  const float* bk       = (const float*)d_in[10];
  const float* Wv       = (const float*)d_in[11];
  const float* bv       = (const float*)d_in[12];
  const float* Wu       = (const float*)d_in[13];
  const float* bu       = (const float*)d_in[14];
  const float* Wt       = (const float*)d_in[15];
  const float* bt       = (const float*)d_in[16];
  const float* osc_u    = (const float*)d_in[17];
  const float* osc_v    = (const float*)d_in[18];

  float* ws     = (float*)d_ws;
  float* at_out = (float*)d_out;
  float* tl_out = at_out + (size_t)Nsz * Dsz;

  kv_kernel   <<<Hsz * Tsz,               128, 0, stream>>>(smoothed, Wk, bk, Wv, bv, ws);
  q_kernel    <<<(Bsz + Tsz) * Hsz,       128, 0, stream>>>(timeslot, upref, user_x, Wq, bq, ws);
  score_kernel<<<21,                      256, 0, stream>>>(ws);
  vw_kernel   <<<Hsz * Tsz,               256, 0, stream>>>(Wu, Wt, ws);
  fused_kernel<<<Nsz / ROWS,              256, 0, stream>>>(hour_x, hour_mask, osc_u, osc_v,
                                                            bu, bt, ws, at_out, tl_out);
}